// Graph_Encoder_85023172592637
// MI455X (gfx1250) — compile-verified
//
#include <hip/hip_runtime.h>

#define V   4
#define T   4
#define NSN 100000
#define NHX 12288
#define EE  100000
#define HH  64

typedef __attribute__((ext_vector_type(2))) float v2f;
typedef __attribute__((ext_vector_type(8))) float v8f;

__device__ __forceinline__ float nan_to_num_f(float x) {
  if (__builtin_isnan(x)) return 0.0f;
  if (__builtin_isinf(x)) return x > 0.0f ? 3.402823466e38f : -3.402823466e38f;
  return x;
}

// ---------------------------------------------------------------- zero scratch
__global__ void zero_ws_kernel(unsigned int* __restrict__ p, int n) {
  int i = blockIdx.x * blockDim.x + threadIdx.x;
  if (i < n) p[i] = 0u;
}

// ---------------------------------------------------------------- degree count
__global__ void degree_kernel(const int* __restrict__ src, const int* __restrict__ dst,
                              unsigned int* __restrict__ deg_out,
                              unsigned int* __restrict__ deg_in) {
  int i = blockIdx.x * blockDim.x + threadIdx.x;
  if (i >= V * T * EE) return;
  int vt = i / EE;                              // v*T + t'
  atomicAdd(&deg_out[(size_t)vt * NSN + src[i]], 1u);
  atomicAdd(&deg_in[vt * NHX + dst[i]], 1u);
}

// ------------------------------------------------- normalized gather + scatter
// s[v,t',t,n] += x[v,t,src] * rsqrt(max(deg_out[v,t',src],1))
__global__ void scatter_kernel(const float* __restrict__ x, const int* __restrict__ src,
                               const int* __restrict__ dst,
                               const unsigned int* __restrict__ deg_out,
                               float* __restrict__ s) {
  int i = blockIdx.x * blockDim.x + threadIdx.x;
  if (i >= V * T * EE) return;
  int vt = i / EE;                              // v*T + t'
  int v  = vt / T;
  int sn = src[i];
  int dn = dst[i];
  float ns = rsqrtf(fmaxf((float)deg_out[(size_t)vt * NSN + sn], 1.0f));
#pragma unroll
  for (int t = 0; t < T; ++t) {
    float xv = nan_to_num_f(x[(size_t)(v * T + t) * NSN + sn]);
    atomicAdd(&s[((size_t)vt * T + t) * NHX + dn], xv * ns);
  }
}

// ---------------------------------------------------------- dst-side normalize
__global__ void norm_dst_kernel(float* __restrict__ s, const unsigned int* __restrict__ deg_in) {
  int i = blockIdx.x * blockDim.x + threadIdx.x;
  if (i >= V * T * T * NHX) return;
  int n   = i % NHX;
  int vtp = i / (T * NHX);                      // v*T + t'
  s[i] *= rsqrtf(fmaxf((float)deg_in[vtp * NHX + n], 1.0f));
}

// ------------------------------------- dense epilogue: rank-1 FP32 WMMA tiles
// out[n,v,t,h] = sum_{t'} leakyrelu(s[v,t',t,n]*W[v,t',h] + b[v,t',h])
// One wave per (v,t,ntile). 16x16x4 f32 WMMA, only K=0 populated, bias carried
// in the C operand (D = A*B + C). All loads are branch-free: every lane loads
// at the clamped index (lane&15) and hi-lanes are zeroed with a select, so
// EXEC stays all-ones around the WMMAs.
__global__ void dense_wmma_kernel(const float* __restrict__ s, const float* __restrict__ W,
                                  const float* __restrict__ b, float* __restrict__ out) {
  const int NT = NHX / 16;                      // 768 n-tiles
  int wave = blockIdx.x * (blockDim.x >> 5) + (threadIdx.x >> 5);
  int lane = threadIdx.x & 31;
  int ntile = wave % NT;
  int vt    = wave / NT;                        // v*T + t
  int t = vt % T;
  int v = vt / T;
  int n0 = ntile * 16;
  int lc = lane & 15;                           // tile column / clamped row index
  bool lo = lane < 16;

  // A (16x4 f32): VGPR0 lanes0-15 = (M=lane, K=0) -> s values; all other K = 0.
  v2f a[T];
#pragma unroll
  for (int tp = 0; tp < T; ++tp) {
    float sv = s[(((size_t)v * T + tp) * T + t) * NHX + n0 + lc];  // all lanes load
    a[tp][0] = lo ? sv : 0.0f;                                     // cndmask, no branch
    a[tp][1] = 0.0f;
  }

#pragma unroll
  for (int ht = 0; ht < HH / 16; ++ht) {
    int h0 = ht * 16;
    v8f acc = {};
#pragma unroll
    for (int tp = 0; tp < T; ++tp) {
      // B (4x16 f32): K=0 row in lanes 0-15 of VGPR0 -> W values; rest zero.
      float wv = W[((size_t)v * T + tp) * HH + h0 + lc];           // all lanes load
      v2f bw;
      bw[0] = lo ? wv : 0.0f;
      bw[1] = 0.0f;
      // C: bias depends only on column N = lane%16 -> splat across all 8 VGPRs.
      float bias = b[((size_t)v * T + tp) * HH + h0 + lc];
      v8f c;
#pragma unroll
      for (int j = 0; j < 8; ++j) c[j] = bias;
      v8f d = __builtin_amdgcn_wmma_f32_16x16x4_f32(
          /*neg_a=*/false, a[tp], /*neg_b=*/false, bw,
          /*c_mod=*/(short)0, c, /*reuse_a=*/false, /*reuse_b=*/false);
#pragma unroll
      for (int j = 0; j < 8; ++j) {
        float y = d[j];
        acc[j] += (y > 0.0f) ? y : 0.01f * y;   // LeakyReLU(0.01)
      }
    }
    // D layout: VGPR j -> row M=j (lanes 0-15) or M=j+8 (lanes 16-31), col = lane%16
    int nrow = n0 + (lo ? 0 : 8);
#pragma unroll
    for (int j = 0; j < 8; ++j) {
      size_t o = (((size_t)(nrow + j) * V + v) * T + t) * HH + (h0 + lc);
      out[o] = acc[j];
    }
  }
}

extern "C" void kernel_launch(void* const* d_in, const int* in_sizes, int n_in,
                              void* d_out, int out_size, void* d_ws, size_t ws_size,
                              hipStream_t stream) {
  const float* x   = (const float*)d_in[0];
  const float* W   = (const float*)d_in[1];
  const float* b   = (const float*)d_in[2];
  const int*   src = (const int*)d_in[3];
  const int*   dst = (const int*)d_in[4];
  float*       out = (float*)d_out;

  // workspace layout (packed 4-byte types)
  unsigned int* deg_out = (unsigned int*)d_ws;                    // [V*T*NSN]
  unsigned int* deg_in  = deg_out + (size_t)V * T * NSN;          // [V*T*NHX]
  float*        s       = (float*)(deg_in + (size_t)V * T * NHX); // [V*T*T*NHX]

  const int zero_words = V * T * NSN + V * T * NHX + V * T * T * NHX;
  zero_ws_kernel<<<(zero_words + 255) / 256, 256, 0, stream>>>((unsigned int*)d_ws, zero_words);

  const int ne = V * T * EE;
  degree_kernel<<<(ne + 255) / 256, 256, 0, stream>>>(src, dst, deg_out, deg_in);
  scatter_kernel<<<(ne + 255) / 256, 256, 0, stream>>>(x, src, dst, deg_out, s);

  const int nsel = V * T * T * NHX;
  norm_dst_kernel<<<(nsel + 255) / 256, 256, 0, stream>>>(s, deg_in);

  const int waves = V * T * (NHX / 16);         // 12288 waves, 4 waves/block
  dense_wmma_kernel<<<waves / 4, 128, 0, stream>>>(s, W, b, out);
}